// DropoutMaxPoolingLayer_45337674777261
// MI455X (gfx1250) — compile-verified
//
#include <hip/hip_runtime.h>
#include <stdint.h>

// Probabilistic weighted 2x2 pooling, NHWC (16,224,224,64) f32 -> (16,112,112,64) f32.
// Pure streaming op (~257MB @ 23.3TB/s => ~11us floor). Uses the CDNA5 async
// copy path (global_load_async_to_lds_b128 + ASYNCcnt split waits) in a
// double-buffered pipeline, NT cache hints for the streamed input/output.

typedef float v4f __attribute__((ext_vector_type(4)));

#define THREADS       256
#define BLOCKS        3136        // 802,816 threads = 1/4 of output vec4s
#define NITER         4
#define ROW_BYTES     57344       // 224*64*4 : one input row
#define IN_ITER_BYTES 51380224    // input quarter per iteration
#define OUT_ITER_EL   3211264     // output quarter (floats) per iteration
#define LDS_BUF_BYTES 16384       // 256 threads * 4 slots * 16B

// Issue the 4 window loads of one iteration straight into this lane's LDS slots.
// GVS form: mem = SGPR_base + VGPR_i32_offset (+imm). The imm offset is added to
// BOTH global and LDS addresses on this instruction, so each load gets its own
// offset VGPR instead. s_wait_dscnt 0 first: the target LDS buffer must have
// been fully consumed by ds_loads before the DMA may overwrite it.
__device__ __forceinline__ void issue4(uint32_t l, int g, uint64_t base) {
  asm volatile(
      "s_wait_dscnt 0\n\t"
      "global_load_async_to_lds_b128 %0, %4, %8 th:TH_LOAD_NT\n\t"
      "global_load_async_to_lds_b128 %1, %5, %8 th:TH_LOAD_NT\n\t"
      "global_load_async_to_lds_b128 %2, %6, %8 th:TH_LOAD_NT\n\t"
      "global_load_async_to_lds_b128 %3, %7, %8 th:TH_LOAD_NT"
      :
      : "v"(l), "v"(l + 16u), "v"(l + 32u), "v"(l + 48u),
        "v"(g), "v"(g + 256), "v"(g + ROW_BYTES), "v"(g + ROW_BYTES + 256),
        "s"(base)
      : "memory");
}

__global__ __launch_bounds__(THREADS) void pwpool_kernel(
    const float* __restrict__ in, float* __restrict__ out) {
  __shared__ v4f lbuf[2][THREADS][4];   // 32KB: double-buffered, 64B per lane

  const int tid = blockIdx.x * THREADS + threadIdx.x;
  const int c4  = tid & 15;             // channel group (4 floats each)
  const int t   = tid >> 4;
  const int w   = t % 112;              // output column
  const int t2  = t / 112;              // fused (batch*112 + out_row)

  // Byte offset of window top-left for iteration 0:
  // elem = 2*t2*14336 + 2*w*64 + 4*c4
  const int gbyte0 = (t2 * 28672 + w * 128 + c4 * 4) * 4;
  const uint64_t sbase = (uint64_t)(uintptr_t)in;

  // Flat(shared) pointer low 32 bits == wave-relative LDS byte address,
  // which is exactly what the async-load VDST operand wants.
  const uint32_t lds0 = (uint32_t)(uintptr_t)&lbuf[0][threadIdx.x][0];

  issue4(lds0, gbyte0, sbase);                                   // iter 0
  issue4(lds0 + LDS_BUF_BYTES, gbyte0 + IN_ITER_BYTES, sbase);   // iter 1

  const int out0 = tid * 4;   // output flat index collapses to 4*tid

#pragma unroll
  for (int i = 0; i < NITER; ++i) {
    // In-order completion: <=4 outstanding means iter i's 4 loads have landed
    // while iter i+1's stay in flight.
    if (i < NITER - 1) asm volatile("s_wait_asynccnt 4" ::: "memory");
    else               asm volatile("s_wait_asynccnt 0" ::: "memory");

    const int buf = i & 1;
    const v4f a = lbuf[buf][threadIdx.x][0];
    const v4f b = lbuf[buf][threadIdx.x][1];
    const v4f c = lbuf[buf][threadIdx.x][2];
    const v4f d = lbuf[buf][threadIdx.x][3];

    // Prefetch iter i+2 into the buffer we just read (issue4 waits dscnt 0).
    if (i + 2 < NITER)
      issue4(lds0 + (uint32_t)(buf * LDS_BUF_BYTES),
             gbyte0 + (i + 2) * IN_ITER_BYTES, sbase);

    // 5-comparator sorting network (per channel), ascending s0..s3.
    const v4f lo0 = __builtin_elementwise_min(a, b);
    const v4f hi0 = __builtin_elementwise_max(a, b);
    const v4f lo1 = __builtin_elementwise_min(c, d);
    const v4f hi1 = __builtin_elementwise_max(c, d);
    const v4f s0  = __builtin_elementwise_min(lo0, lo1);
    const v4f tm  = __builtin_elementwise_max(lo0, lo1);
    const v4f s3  = __builtin_elementwise_max(hi0, hi1);
    const v4f um  = __builtin_elementwise_min(hi0, hi1);
    const v4f s1  = __builtin_elementwise_min(tm, um);
    const v4f s2  = __builtin_elementwise_max(tm, um);

    // probs[k] = 0.5 * 0.5^(3-k) = {0.0625, 0.125, 0.25, 0.5}
    const v4f r = 0.0625f * s0 + 0.125f * s1 + 0.25f * s2 + 0.5f * s3;

    __builtin_nontemporal_store(r, (v4f*)(out + out0 + i * OUT_ITER_EL));
  }
}

extern "C" void kernel_launch(void* const* d_in, const int* in_sizes, int n_in,
                              void* d_out, int out_size, void* d_ws, size_t ws_size,
                              hipStream_t stream) {
  (void)in_sizes; (void)n_in; (void)out_size; (void)d_ws; (void)ws_size;
  const float* in = (const float*)d_in[0];
  float* out = (float*)d_out;
  pwpool_kernel<<<BLOCKS, THREADS, 0, stream>>>(in, out);
}